// BFS_Neural_Execution_39213051412852
// MI455X (gfx1250) — compile-verified
//
#include <hip/hip_runtime.h>
#include <hip/hip_bf16.h>
#include <stdint.h>

#define HDIM 128

typedef float v2f __attribute__((ext_vector_type(2)));
typedef float v8f __attribute__((ext_vector_type(8)));

// ---------------------------------------------------------------------------
// Native fp32 WMMA: D(16x16) += A(16x4) * B(4x16), wave32.
// A frag: lane L -> row M=L%16, K pair starting at 2*(L/16)
// B frag: lane L -> col N=L%16, VGPR0=B[2*(L/16)][N], VGPR1=B[2*(L/16)+1][N]
// C/D:    VGPR r, lane L -> M = r + 8*(L/16), N = L%16
// ---------------------------------------------------------------------------
__device__ __forceinline__ v8f wmma_f32_k4(v2f a, v2f b, v8f c) {
    return __builtin_amdgcn_wmma_f32_16x16x4_f32(
        /*neg_a=*/false, a, /*neg_b=*/false, b,
        /*c_mod=*/(short)0, c, /*reuse_a=*/false, /*reuse_b=*/false);
}

// LDS row strides (floats): pad so row*stride mod 64 banks = 4*row (conflict-free b64 reads)
#define STRIDE128 132   // K=128 tiles
#define STRIDE256 260   // K=256 tiles

// Chunked, ping-pong A-fragment GEMM body: KSTEPS k-steps of 4, 8 steps/chunk.
// Loads for chunk c+1 are issued before the WMMAs of chunk c, so the LDS
// latency hides under the matrix ops (wait becomes dscnt<=4, not 0).
template <int KSTEPS>
__device__ __forceinline__ v8f gemm_body(const float* __restrict__ arow,
                                         const v2f* __restrict__ bf) {
    v8f cacc = {};
    v2f ab[2][8];
#pragma unroll
    for (int j = 0; j < 8; ++j)
        ab[0][j] = *(const v2f*)(arow + 4 * j);
#pragma unroll
    for (int kc = 0; kc < KSTEPS; kc += 8) {
        const int cur = (kc >> 3) & 1;
        const int nxt = cur ^ 1;
        if (kc + 8 < KSTEPS) {
#pragma unroll
            for (int j = 0; j < 8; ++j)
                ab[nxt][j] = *(const v2f*)(arow + 4 * (kc + 8 + j));
        }
#pragma unroll
        for (int j = 0; j < 8; ++j)
            cacc = wmma_f32_k4(ab[cur][j], bf[kc + j], cacc);
    }
    return cacc;
}

// ---------------------------------------------------------------------------
// zero-fill workspace region (agg buffer must start at 0.0f every call)
// ---------------------------------------------------------------------------
__global__ void zero_kernel(float* __restrict__ p, long long n) {
    long long i = (long long)blockIdx.x * blockDim.x + threadIdx.x;
    if (i < n) p[i] = 0.0f;
}

// ---------------------------------------------------------------------------
// Encoder: z = relu(pre_h @ W_enc[1:,:] + x * W_enc[0,:] + b_enc)
// block = 256 (8 waves = 8 N-tiles), each block processes ENC_MTB M-tiles.
// A staged in LDS (prefetched one tile ahead into registers);
// B fragments persistent in registers.
// ---------------------------------------------------------------------------
#define ENC_MTB 5
__global__ __launch_bounds__(256, 1)
void encoder_kernel(const float* __restrict__ x,
                    const float* __restrict__ pre_h,
                    const float* __restrict__ W_enc,  // (129,128)
                    const float* __restrict__ b_enc,
                    float* __restrict__ z, int N) {
    __shared__ float As[16 * STRIDE128];

    const int tid   = threadIdx.x;
    const int wave  = tid >> 5;
    const int lane  = tid & 31;
    const int mloc  = lane & 15;
    const int khalf = lane >> 4;
    const int nl    = lane & 15;
    const int n     = wave * 16 + nl;

    // persistent B fragments: W_enc rows 1..128, column n
    v2f bf[32];
    {
        const float* bcol = W_enc + HDIM + n;   // skip x-coefficient row 0
#pragma unroll
        for (int k = 0; k < 32; ++k) {
            const int kr = 4 * k + 2 * khalf;
            bf[k][0] = bcol[(size_t)kr * HDIM];
            bf[k][1] = bcol[(size_t)(kr + 1) * HDIM];
        }
    }
    const float w0 = W_enc[n];
    const float bb = b_enc[n];

    // staging geometry: thread -> row tid/16, 8 floats at col (tid%16)*8
    const int sr  = tid >> 4;
    const int sc0 = (tid & 15) * 8;
    float*    dp  = As + sr * STRIDE128 + sc0;
    const int mt0 = blockIdx.x * ENC_MTB;

    // prologue: stage tile 0
    float4 g0, g1;
    {
        const float* sp = pre_h + (size_t)(mt0 * 16 + sr) * HDIM + sc0;
        g0 = *(const float4*)(sp);
        g1 = *(const float4*)(sp + 4);
    }
    *(float4*)(dp)     = g0;
    *(float4*)(dp + 4) = g1;
    __syncthreads();

#pragma unroll 1
    for (int t = 0; t < ENC_MTB; ++t) {
        const int mt = mt0 + t;
        // prefetch next tile's staging data into registers (overlaps compute)
        if (t + 1 < ENC_MTB) {
            const float* sp = pre_h + (size_t)((mt + 1) * 16 + sr) * HDIM + sc0;
            g0 = *(const float4*)(sp);
            g1 = *(const float4*)(sp + 4);
        }

        const float* arow = As + mloc * STRIDE128 + 2 * khalf;
        v8f cacc = gemm_body<32>(arow, bf);

#pragma unroll
        for (int r = 0; r < 8; ++r) {
            const int rr = mt * 16 + r + 8 * khalf;
            float v = cacc[r] + x[rr] * w0 + bb;
            z[(size_t)rr * HDIM + n] = fmaxf(v, 0.0f);
        }

        __syncthreads();                 // all waves done reading As
        if (t + 1 < ENC_MTB) {
            *(float4*)(dp)     = g0;     // store next tile
            *(float4*)(dp + 4) = g1;
            __syncthreads();
        }
    }
}

// ---------------------------------------------------------------------------
// Message + scatter-max:
//   m = relu([z[dst] | z[src] | attr] @ W_M + b_M)
//   agg[dst] = max(agg[dst], m)   via u32 atomicMax on float bits (m >= 0)
// Concatenated K=256 A-tile in LDS; B frags (64 x v2f) persistent in registers.
// ---------------------------------------------------------------------------
#define MSG_MTB 4
__global__ __launch_bounds__(256, 1)
void message_kernel(const float* __restrict__ z,
                    const int*   __restrict__ edge_index, // (2,E)
                    const float* __restrict__ edge_attr,
                    const float* __restrict__ W_M,        // (257,128)
                    const float* __restrict__ b_M,
                    unsigned int* __restrict__ agg_bits,  // (N,128)
                    int E) {
    __shared__ float As[16 * STRIDE256];

    const int tid   = threadIdx.x;
    const int wave  = tid >> 5;
    const int lane  = tid & 31;
    const int mloc  = lane & 15;
    const int khalf = lane >> 4;
    const int nl    = lane & 15;
    const int n     = wave * 16 + nl;

    // persistent B fragments: W_M rows 0..255, column n
    v2f bf[64];
    {
        const float* bcol = W_M + n;
#pragma unroll
        for (int k = 0; k < 64; ++k) {
            const int kr = 4 * k + 2 * khalf;
            bf[k][0] = bcol[(size_t)kr * HDIM];
            bf[k][1] = bcol[(size_t)(kr + 1) * HDIM];
        }
    }
    const float wa = W_M[(size_t)256 * HDIM + n];
    const float bb = b_M[n];

    // staging geometry: thread -> row tid/16, 16 floats at col (tid%16)*16
    const int sr  = tid >> 4;
    const int sc0 = (tid & 15) * 16;
    float*    dp  = As + sr * STRIDE256 + sc0;
    const int et0 = blockIdx.x * MSG_MTB;

    // prologue: stage tile 0  ([z_dst | z_src] concatenated along K)
    float4 g0, g1, g2, g3;
    {
        const int e = et0 * 16 + sr;
        const int s = edge_index[e];
        const int d = edge_index[E + e];
        const float* sp = (sc0 < HDIM) ? (z + (size_t)d * HDIM + sc0)
                                       : (z + (size_t)s * HDIM + (sc0 - HDIM));
        g0 = *(const float4*)(sp);
        g1 = *(const float4*)(sp + 4);
        g2 = *(const float4*)(sp + 8);
        g3 = *(const float4*)(sp + 12);
    }
    *(float4*)(dp)      = g0;
    *(float4*)(dp + 4)  = g1;
    *(float4*)(dp + 8)  = g2;
    *(float4*)(dp + 12) = g3;
    __syncthreads();

#pragma unroll 1
    for (int t = 0; t < MSG_MTB; ++t) {
        const int et = et0 + t;
        if (t + 1 < MSG_MTB) {
            const int e = (et + 1) * 16 + sr;
            const int s = edge_index[e];
            const int d = edge_index[E + e];
            const float* sp = (sc0 < HDIM) ? (z + (size_t)d * HDIM + sc0)
                                           : (z + (size_t)s * HDIM + (sc0 - HDIM));
            g0 = *(const float4*)(sp);
            g1 = *(const float4*)(sp + 4);
            g2 = *(const float4*)(sp + 8);
            g3 = *(const float4*)(sp + 12);
        }

        const float* arow = As + mloc * STRIDE256 + 2 * khalf;
        v8f cacc = gemm_body<64>(arow, bf);

#pragma unroll
        for (int r = 0; r < 8; ++r) {
            const int ee = et * 16 + r + 8 * khalf;
            const int dd = edge_index[E + ee];
            float v = fmaxf(cacc[r] + edge_attr[ee] * wa + bb, 0.0f);
            // v >= 0: float order == unsigned order of bit patterns
            atomicMax(&agg_bits[(size_t)dd * HDIM + n], __float_as_uint(v));
        }

        __syncthreads();
        if (t + 1 < MSG_MTB) {
            *(float4*)(dp)      = g0;
            *(float4*)(dp + 4)  = g1;
            *(float4*)(dp + 8)  = g2;
            *(float4*)(dp + 12) = g3;
            __syncthreads();
        }
    }
}

// ---------------------------------------------------------------------------
// Update: h = relu([z | agg] @ W_U + b_U)   (K=256, contiguous rows)
// ---------------------------------------------------------------------------
#define UPD_MTB 5
__global__ __launch_bounds__(256, 1)
void update_kernel(const float* __restrict__ z,
                   const float* __restrict__ agg,
                   const float* __restrict__ W_U,  // (256,128)
                   const float* __restrict__ b_U,
                   float* __restrict__ h, int N) {
    __shared__ float As[16 * STRIDE256];

    const int tid   = threadIdx.x;
    const int wave  = tid >> 5;
    const int lane  = tid & 31;
    const int mloc  = lane & 15;
    const int khalf = lane >> 4;
    const int nl    = lane & 15;
    const int n     = wave * 16 + nl;

    v2f bf[64];
    {
        const float* bcol = W_U + n;
#pragma unroll
        for (int k = 0; k < 64; ++k) {
            const int kr = 4 * k + 2 * khalf;
            bf[k][0] = bcol[(size_t)kr * HDIM];
            bf[k][1] = bcol[(size_t)(kr + 1) * HDIM];
        }
    }
    const float bb = b_U[n];

    const int sr  = tid >> 4;
    const int sc0 = (tid & 15) * 16;
    float*    dp  = As + sr * STRIDE256 + sc0;
    const int mt0 = blockIdx.x * UPD_MTB;

    float4 g0, g1, g2, g3;
    {
        const int rr = mt0 * 16 + sr;
        const float* sp = (sc0 < HDIM) ? (z   + (size_t)rr * HDIM + sc0)
                                       : (agg + (size_t)rr * HDIM + (sc0 - HDIM));
        g0 = *(const float4*)(sp);
        g1 = *(const float4*)(sp + 4);
        g2 = *(const float4*)(sp + 8);
        g3 = *(const float4*)(sp + 12);
    }
    *(float4*)(dp)      = g0;
    *(float4*)(dp + 4)  = g1;
    *(float4*)(dp + 8)  = g2;
    *(float4*)(dp + 12) = g3;
    __syncthreads();

#pragma unroll 1
    for (int t = 0; t < UPD_MTB; ++t) {
        const int mt = mt0 + t;
        if (t + 1 < UPD_MTB) {
            const int rr = (mt + 1) * 16 + sr;
            const float* sp = (sc0 < HDIM) ? (z   + (size_t)rr * HDIM + sc0)
                                           : (agg + (size_t)rr * HDIM + (sc0 - HDIM));
            g0 = *(const float4*)(sp);
            g1 = *(const float4*)(sp + 4);
            g2 = *(const float4*)(sp + 8);
            g3 = *(const float4*)(sp + 12);
        }

        const float* arow = As + mloc * STRIDE256 + 2 * khalf;
        v8f cacc = gemm_body<64>(arow, bf);

#pragma unroll
        for (int r = 0; r < 8; ++r) {
            const int rr = mt * 16 + r + 8 * khalf;
            h[(size_t)rr * HDIM + n] = fmaxf(cacc[r] + bb, 0.0f);
        }

        __syncthreads();
        if (t + 1 < UPD_MTB) {
            *(float4*)(dp)      = g0;
            *(float4*)(dp + 4)  = g1;
            *(float4*)(dp + 8)  = g2;
            *(float4*)(dp + 12) = g3;
            __syncthreads();
        }
    }
}

// ---------------------------------------------------------------------------
// Decoder: y[i] = z[i].W_dec[0:128] + h[i].W_dec[128:256] + b_dec
// one wave per node; coalesced lane-strided dot + wave reduction
// ---------------------------------------------------------------------------
__global__ void decoder_kernel(const float* __restrict__ z,
                               const float* __restrict__ h,
                               const float* __restrict__ W_dec, // (256,1)
                               const float* __restrict__ b_dec,
                               float* __restrict__ y, int N) {
    const int gw   = (int)((blockIdx.x * (long long)blockDim.x + threadIdx.x) >> 5);
    const int lane = threadIdx.x & 31;
    if (gw >= N) return;
    const float* zr = z + (size_t)gw * HDIM;
    const float* hr = h + (size_t)gw * HDIM;
    float acc = 0.0f;
#pragma unroll
    for (int k = lane; k < HDIM; k += 32)
        acc = fmaf(zr[k], W_dec[k], fmaf(hr[k], W_dec[HDIM + k], acc));
#pragma unroll
    for (int off = 16; off > 0; off >>= 1)
        acc += __shfl_xor(acc, off, 32);
    if (lane == 0) y[gw] = acc + b_dec[0];
}

// ---------------------------------------------------------------------------
// Deterministic column mean of h: one block per column (128 blocks)
// ---------------------------------------------------------------------------
__global__ void colmean_kernel(const float* __restrict__ h,
                               float* __restrict__ h_mean, int N) {
    __shared__ float sm[256];
    const int col = blockIdx.x;
    float acc = 0.0f;
    for (int row = threadIdx.x; row < N; row += 256)
        acc += h[(size_t)row * HDIM + col];
    sm[threadIdx.x] = acc;
    __syncthreads();
    for (int s = 128; s > 0; s >>= 1) {
        if ((int)threadIdx.x < s) sm[threadIdx.x] += sm[threadIdx.x + s];
        __syncthreads();
    }
    if (threadIdx.x == 0) h_mean[col] = sm[0] / (float)N;
}

// ---------------------------------------------------------------------------
// tau = h_mean . (W_term[0:128] + W_term[128:256]) + b_term
// ---------------------------------------------------------------------------
__global__ void tau_kernel(const float* __restrict__ h_mean,
                           const float* __restrict__ W_term, // (256,1)
                           const float* __restrict__ b_term,
                           float* __restrict__ tau_out) {
    __shared__ float sm[128];
    const int t = threadIdx.x;
    sm[t] = h_mean[t] * (W_term[t] + W_term[HDIM + t]);
    __syncthreads();
    for (int s = 64; s > 0; s >>= 1) {
        if (t < s) sm[t] += sm[t + s];
        __syncthreads();
    }
    if (t == 0) tau_out[0] = sm[0] + b_term[0];
}

// ---------------------------------------------------------------------------
extern "C" void kernel_launch(void* const* d_in, const int* in_sizes, int n_in,
                              void* d_out, int out_size, void* d_ws, size_t ws_size,
                              hipStream_t stream) {
    const float* x          = (const float*)d_in[0];   // (N,1)
    const float* pre_h      = (const float*)d_in[1];   // (N,H)
    const int*   edge_index = (const int*)  d_in[2];   // (2,E)
    const float* edge_attr  = (const float*)d_in[3];   // (E,1)
    const float* W_enc      = (const float*)d_in[4];   // (129,128)
    const float* b_enc      = (const float*)d_in[5];
    const float* W_M        = (const float*)d_in[6];   // (257,128)
    const float* b_M        = (const float*)d_in[7];
    const float* W_U        = (const float*)d_in[8];   // (256,128)
    const float* b_U        = (const float*)d_in[9];
    const float* W_dec      = (const float*)d_in[10];  // (256,1)
    const float* b_dec      = (const float*)d_in[11];
    const float* W_term     = (const float*)d_in[12];  // (256,1)
    const float* b_term     = (const float*)d_in[13];

    const int N = in_sizes[0];        // 50000 = 16 * 3125, 3125 = 5 * 625
    const int E = in_sizes[3];        // 640000 = 16 * 40000, 40000 = 4 * 10000

    // outputs: h (N*H) | y (N) | tau (1)
    float* h_out   = (float*)d_out;
    float* y_out   = h_out + (size_t)N * HDIM;
    float* tau_out = y_out + N;

    // workspace: z (N*H) | agg (N*H) | h_mean (128)
    float* z      = (float*)d_ws;
    float* agg    = z + (size_t)N * HDIM;
    float* h_mean = agg + (size_t)N * HDIM;

    const long long aggN = (long long)N * HDIM;

    // 1) agg := 0 (required for bitwise float max; ws is not re-zeroed by harness)
    zero_kernel<<<(unsigned)((aggN + 255) / 256), 256, 0, stream>>>(agg, aggN);

    // 2) encoder
    encoder_kernel<<<N / 16 / ENC_MTB, 256, 0, stream>>>(x, pre_h, W_enc, b_enc, z, N);

    // 3) messages + segment max
    message_kernel<<<E / 16 / MSG_MTB, 256, 0, stream>>>(z, edge_index, edge_attr,
                                                         W_M, b_M, (unsigned int*)agg, E);

    // 4) update -> h (directly into d_out)
    update_kernel<<<N / 16 / UPD_MTB, 256, 0, stream>>>(z, agg, W_U, b_U, h_out, N);

    // 5) decoder -> y
    {
        long long threads = (long long)N * 32;
        decoder_kernel<<<(unsigned)((threads + 255) / 256), 256, 0, stream>>>(
            z, h_out, W_dec, b_dec, y_out, N);
    }

    // 6) terminator -> tau
    colmean_kernel<<<HDIM, 256, 0, stream>>>(h_out, h_mean, N);
    tau_kernel<<<1, 128, 0, stream>>>(h_mean, W_term, b_term, tau_out);
}